// FCGF_MLP2_89575837925683
// MI455X (gfx1250) — compile-verified
//
#include <hip/hip_runtime.h>

#define B_SEG 4096
#define C_IN  32
#define C_OUT 128
#define BN_EPS 1e-5f
#define NEG_FLT_MAX (-3.402823466e+38f)

typedef __attribute__((ext_vector_type(2))) float v2f;
typedef __attribute__((ext_vector_type(8))) float v8f;

// ---------------------------------------------------------------------------
// Kernel 0: exclusive prefix sum of segment lengths (4096 ints, one block)
// ---------------------------------------------------------------------------
__global__ __launch_bounds__(1024) void seg_offsets_kernel(
    const int* __restrict__ len, int* __restrict__ off) {
  __shared__ int part[1024];
  const int t = threadIdx.x;
  const int l0 = len[t * 4 + 0];
  const int l1 = len[t * 4 + 1];
  const int l2 = len[t * 4 + 2];
  const int l3 = len[t * 4 + 3];
  part[t] = l0 + l1 + l2 + l3;
  __syncthreads();
  // Hillis-Steele inclusive scan over the 1024 partial sums
  for (int d = 1; d < 1024; d <<= 1) {
    int v = (t >= d) ? part[t - d] : 0;
    __syncthreads();
    part[t] += v;
    __syncthreads();
  }
  const int base = (t == 0) ? 0 : part[t - 1];
  off[t * 4 + 0] = base;
  off[t * 4 + 1] = base + l0;
  off[t * 4 + 2] = base + l0 + l1;
  off[t * 4 + 3] = base + l0 + l1 + l2;
}

// ---------------------------------------------------------------------------
// Kernel 1: ragged segment max-pool. One block per segment.
// lane (0..31) = channel -> each row read is one coalesced 128B line.
// 8 waves stride over rows; cross-wave reduce through LDS.
// ---------------------------------------------------------------------------
__global__ __launch_bounds__(256) void seg_maxpool_kernel(
    const float* __restrict__ x, const int* __restrict__ len,
    const int* __restrict__ off, float* __restrict__ pooled) {
  __shared__ float red[8][C_IN];
  const int b    = blockIdx.x;
  const int lane = threadIdx.x & 31;
  const int wave = threadIdx.x >> 5;
  const int start = off[b];
  const int L     = len[b];
  const float* xp = x + (size_t)start * C_IN + lane;

  float m = NEG_FLT_MAX;
  for (int r = wave; r < L; r += 8) {
    // speculative prefetch ~64 rows ahead (gfx1250 global_prefetch_b8)
    __builtin_prefetch(xp + (size_t)(r + 64) * C_IN, 0, 0);
    m = fmaxf(m, xp[(size_t)r * C_IN]);
  }
  red[wave][lane] = m;
  __syncthreads();
  if (wave == 0) {
    float mm = red[0][lane];
#pragma unroll
    for (int w = 1; w < 8; ++w) mm = fmaxf(mm, red[w][lane]);
    pooled[(size_t)b * C_IN + lane] = mm;
  }
}

// ---------------------------------------------------------------------------
// Kernel 2: y[4096,128] = pooled[4096,32] @ conv_w^T[32,128] + bias
// One wave per 16x16 tile, V_WMMA_F32_16X16X4_F32, 8 k-steps (K=32).
//
// A layout (16x4 f32, ISA 7.12.2): lane L holds row M=L%16;
//   VGPR v holds K = 2*(L/16) + v  -> contiguous float2 per k-step.
// B layout (4x16 f32, symmetric striping): lane L holds col N=L%16;
//   VGPR v holds K = 2*(L/16) + v. conv_w is [C_OUT, C_IN] row-major, and
//   B[k][n] = conv_w[n][k], so each lane's fragment is a contiguous float2
//   from conv_w row (n0 + L%16).
// C/D layout: element (M = vgpr + 8*(L/16), N = L%16).
// ---------------------------------------------------------------------------
__global__ __launch_bounds__(128) void gemm_wmma_kernel(
    const float* __restrict__ pooled, const float* __restrict__ w,
    const float* __restrict__ bias, float* __restrict__ y) {
  const int lane   = threadIdx.x & 31;
  const int waveId = blockIdx.x * 4 + (threadIdx.x >> 5);
  const int tm = waveId >> 3;        // 0..255  (M tiles)
  const int tn = waveId & 7;         // 0..7    (N tiles)
  const int row  = lane & 15;
  const int half = lane >> 4;
  const int m0 = tm * 16, n0 = tn * 16;

  v8f c;
  const float bv = bias[n0 + row];
#pragma unroll
  for (int i = 0; i < 8; ++i) c[i] = bv;

  const float* arow = pooled + (size_t)(m0 + row) * C_IN;
  const float* brow = w + (size_t)(n0 + row) * C_IN;

#pragma unroll
  for (int s = 0; s < 8; ++s) {
    const int kbase = s * 4 + half * 2;
    v2f a = *(const v2f*)(arow + kbase);
    v2f b = *(const v2f*)(brow + kbase);
    // (neg_a, A, neg_b, B, c_mod, C, reuse_a, reuse_b)
    c = __builtin_amdgcn_wmma_f32_16x16x4_f32(false, a, false, b,
                                              (short)0, c, false, false);
  }

#pragma unroll
  for (int i = 0; i < 8; ++i) {
    y[(size_t)(m0 + i + 8 * half) * C_OUT + (n0 + row)] = c[i];
  }
}

// ---------------------------------------------------------------------------
// Kernel 3: per-channel mean & biased var over the batch dim (one block/chan)
// ---------------------------------------------------------------------------
__global__ __launch_bounds__(256) void bn_stats_kernel(
    const float* __restrict__ y, float* __restrict__ stats) {
  __shared__ float ssum[256];
  __shared__ float ssq[256];
  const int c = blockIdx.x;
  const int t = threadIdx.x;
  float sum = 0.f, sq = 0.f;
  for (int r = t; r < B_SEG; r += 256) {
    const float v = y[(size_t)r * C_OUT + c];
    sum += v;
    sq += v * v;
  }
  ssum[t] = sum;
  ssq[t] = sq;
  __syncthreads();
  for (int d = 128; d > 0; d >>= 1) {
    if (t < d) {
      ssum[t] += ssum[t + d];
      ssq[t] += ssq[t + d];
    }
    __syncthreads();
  }
  if (t == 0) {
    const float mean = ssum[0] * (1.0f / B_SEG);
    const float var  = ssq[0] * (1.0f / B_SEG) - mean * mean;
    stats[c]         = mean;
    stats[C_OUT + c] = var;
  }
}

// ---------------------------------------------------------------------------
// Kernel 4: in-place BatchNorm affine + ReLU on d_out
// ---------------------------------------------------------------------------
__global__ __launch_bounds__(256) void bn_relu_kernel(
    float* __restrict__ y, const float* __restrict__ stats,
    const float* __restrict__ gamma, const float* __restrict__ beta) {
  const int i = blockIdx.x * 256 + threadIdx.x;
  if (i >= B_SEG * C_OUT) return;
  const int c = i & (C_OUT - 1);
  const float mean = stats[c];
  const float var  = stats[C_OUT + c];
  float v = (y[i] - mean) * rsqrtf(var + BN_EPS) * gamma[c] + beta[c];
  y[i] = v > 0.f ? v : 0.f;
}

// ---------------------------------------------------------------------------
extern "C" void kernel_launch(void* const* d_in, const int* in_sizes, int n_in,
                              void* d_out, int out_size, void* d_ws,
                              size_t ws_size, hipStream_t stream) {
  const float* x      = (const float*)d_in[0];
  const int*   length = (const int*)d_in[1];
  const float* conv_w = (const float*)d_in[2];
  const float* conv_b = (const float*)d_in[3];
  const float* gamma  = (const float*)d_in[4];
  const float* beta   = (const float*)d_in[5];
  float* y = (float*)d_out;  // [4096, 128], BN+ReLU applied in place

  char* ws = (char*)d_ws;
  int*   offsets = (int*)ws;                                   // 16 KB
  float* pooled  = (float*)(ws + 16 * 1024);                   // 512 KB
  float* stats   = (float*)(ws + 16 * 1024 + (size_t)B_SEG * C_IN * sizeof(float));  // 1 KB

  seg_offsets_kernel<<<1, 1024, 0, stream>>>(length, offsets);
  seg_maxpool_kernel<<<B_SEG, 256, 0, stream>>>(x, length, offsets, pooled);
  gemm_wmma_kernel<<<(B_SEG / 16) * (C_OUT / 16) / 4, 128, 0, stream>>>(
      pooled, conv_w, conv_b, y);
  bn_stats_kernel<<<C_OUT, 256, 0, stream>>>(y, stats);
  bn_relu_kernel<<<(B_SEG * C_OUT) / 256, 256, 0, stream>>>(y, stats, gamma,
                                                            beta);
}